// CMUHLayer1_52286931862213
// MI455X (gfx1250) — compile-verified
//
#include <hip/hip_runtime.h>

// ---------------------------------------------------------------------------
// Grouped conv1d (K=2, 16 groups, 64 nodes) + leaky-ReLU, fp32.
// Roofline: 2.85 GFLOP vs ~550 MB traffic -> output-stream bound (~24us floor
// at 23.3 TB/s). Strategy: per-group conv as GEMM on the fp32 WMMA pipe
// (v_wmma_f32_16x16x4_f32), x-tile staged in LDS (coalesced in, ds-gather
// out), zero-padded B panels register-resident, NT stores for the 512MB out.
// ---------------------------------------------------------------------------

#define T_LEN   4096
#define B_SZ    32
#define CH_IN   70
#define NODE    64
#define NGROUP  16
#define KPAD    20   // max 2*cin = 18 -> 5 K-steps of 4, zero-padded panel rows
#define TT      4    // 16-row t-tiles per wave
#define TROWS   (TT * 16)   // 64 t rows per block
#define HALO    1           // k=1 tap needs one extra row

typedef float v2f __attribute__((ext_vector_type(2)));
typedef float v8f __attribute__((ext_vector_type(8)));

__constant__ int GSIZE[NGROUP] = {9,6,6,6,6,6,5,5,5,5,5,5,5,3,4,4};
__constant__ int GOFF [NGROUP] = {0,9,15,21,27,33,39,44,49,54,59,64,69,74,77,81};
__constant__ int GDATA[85] = {
    0,1,2,3,4,5,26,27,28,
    26,27,28,29,30,31,
    29,30,31,32,33,34,
    32,33,34,35,36,37,
    35,36,37,38,39,40,
    38,39,40,41,42,43,
    6,7,8,9,10,
    9,10,11,12,13,
    11,12,13,14,15,
    16,17,18,19,20,
    19,20,21,22,23,
    21,22,23,24,25,
    44,45,46,47,48,
    47,48,49,
    49,50,51,52,
    50,51,52,53};

struct WPtrs { const float* W[NGROUP]; const float* b[NGROUP]; };

// Pack each group's (2,cin,64) weights into a zero-padded (KPAD x 64) B-panel
// plus bias; W[k,c,n] flattens row-major to exactly B[kk=k*cin+c][n].
__global__ void pack_weights(WPtrs p, float* __restrict__ wsB,
                             float* __restrict__ wsBias) {
    const int g  = blockIdx.x;
    const int n  = threadIdx.x;                 // 64 threads
    const int K2 = 2 * GSIZE[g];
    const float* W = p.W[g];
    for (int kk = 0; kk < KPAD; ++kk)
        wsB[(size_t)(g * KPAD + kk) * NODE + n] = (kk < K2) ? W[kk * NODE + n] : 0.0f;
    wsBias[g * NODE + n] = p.b[g][n];
}

__global__ __launch_bounds__(256) void conv_wmma(const float* __restrict__ x,
                                                 const float* __restrict__ wsB,
                                                 const float* __restrict__ wsBias,
                                                 float* __restrict__ out) {
    __shared__ float xs[(TROWS + HALO) * CH_IN];   // 65*70*4 = 18.2 KB

    const int tid  = threadIdx.x;
    const int wave = tid >> 5;                  // 0..7
    const int lane = tid & 31;
    const int half = lane >> 4;                 // 0/1: K-pair / C-row-half select
    const int m    = lane & 15;                 // A row within t-tile
    const int n15  = lane & 15;                 // B/C column within n-tile
    const int bb   = blockIdx.y;
    const int t0B  = blockIdx.x * TROWS;

    // ---- Stage x tile: coalesced global -> LDS; zero-fill the T boundary ----
    for (int i = tid; i < (TROWS + HALO) * CH_IN; i += 256) {
        const int row = i / CH_IN;
        const int ch  = i - row * CH_IN;
        const int t   = t0B + row;
        xs[i] = (t < T_LEN) ? x[((size_t)bb * T_LEN + t) * CH_IN + ch] : 0.0f;
    }
    __syncthreads();

    for (int gi = 0; gi < 2; ++gi) {
        const int g    = wave + gi * 8;
        const int cin  = GSIZE[g];
        const int K2   = 2 * cin;
        const int ks   = (K2 + 3) >> 2;         // wave-uniform K-step count
        const int goff = GOFF[g];

        // Per-lane LDS gather offsets, invariant across t-tiles: for K-index
        // j = 4s + 2*half + w, element = xs[row*70 + k*70 + chan], k = j/cin.
        int offA[5][2];
#pragma unroll
        for (int s = 0; s < 5; ++s) {
#pragma unroll
            for (int w = 0; w < 2; ++w) {
                const int j = 4 * s + 2 * half + w;
                if (j < K2) {
                    const int k = (j >= cin) ? 1 : 0;
                    const int c = j - (k ? cin : 0);
                    offA[s][w] = k * CH_IN + GDATA[goff + c];
                } else {
                    offA[s][w] = -1;            // B panel row is zero anyway
                }
            }
        }

        // B fragments, register-resident across the TT t-tiles.
        // 4x16 f32 B layout: v0 = row jb, v1 = row jb+1; jb = 4s + 2*half.
        v2f Bf[5][4];
#pragma unroll
        for (int s = 0; s < 5; ++s) {
            const int jb = 4 * s + 2 * half;
#pragma unroll
            for (int nt = 0; nt < 4; ++nt) {
                const float* bp = wsB + (size_t)(g * KPAD + jb) * NODE + nt * 16 + n15;
                Bf[s][nt].x = bp[0];
                Bf[s][nt].y = bp[NODE];
            }
        }
        float bias[4];
#pragma unroll
        for (int nt = 0; nt < 4; ++nt)
            bias[nt] = wsBias[g * NODE + nt * 16 + n15];

        for (int tt = 0; tt < TT; ++tt) {
            const int rbase = (tt * 16 + m) * CH_IN;   // lane's A row in LDS

            v8f acc[4];
#pragma unroll
            for (int nt = 0; nt < 4; ++nt)
#pragma unroll
                for (int r = 0; r < 8; ++r) acc[nt][r] = bias[nt];

#pragma unroll
            for (int s = 0; s < 5; ++s) {
                if (s >= ks) break;             // scalar-uniform: EXEC full at WMMA
                const int o0 = offA[s][0];
                const int o1 = offA[s][1];
                v2f a;
                a.x = (o0 >= 0) ? xs[rbase + o0] : 0.0f;
                a.y = (o1 >= 0) ? xs[rbase + o1] : 0.0f;
#pragma unroll
                for (int nt = 0; nt < 4; ++nt)
                    acc[nt] = __builtin_amdgcn_wmma_f32_16x16x4_f32(
                        /*neg_a=*/false, a, /*neg_b=*/false, Bf[s][nt],
                        /*c_mod=*/(short)0, acc[nt],
                        /*reuse_a=*/false, /*reuse_b=*/false);
            }

            // C/D layout: lanes 0-15 rows r, lanes 16-31 rows r+8.
            // Leaky-ReLU then non-temporal store (512MB stream; keep L2 for x).
#pragma unroll
            for (int nt = 0; nt < 4; ++nt) {
#pragma unroll
                for (int r = 0; r < 8; ++r) {
                    float v = acc[nt][r];
                    v = fmaxf(v, 0.2f * v);
                    const int t = t0B + tt * 16 + r + half * 8;
                    const size_t o = ((size_t)bb * T_LEN + t) * (NODE * NGROUP)
                                   + (size_t)g * NODE + nt * 16 + n15;
                    __builtin_nontemporal_store(v, &out[o]);
                }
            }
        }
    }
}

extern "C" void kernel_launch(void* const* d_in, const int* in_sizes, int n_in,
                              void* d_out, int out_size, void* d_ws, size_t ws_size,
                              hipStream_t stream) {
    (void)in_sizes; (void)n_in; (void)out_size; (void)ws_size;

    const float* x = (const float*)d_in[0];
    WPtrs p;
    for (int i = 0; i < NGROUP; ++i) {
        p.W[i] = (const float*)d_in[1 + i];           // Ws[0..15]
        p.b[i] = (const float*)d_in[1 + NGROUP + i];  // bs[0..15]
    }

    float* wsB    = (float*)d_ws;                     // 16*20*64 floats = 80KB
    float* wsBias = wsB + NGROUP * KPAD * NODE;       // +16*64 floats

    pack_weights<<<NGROUP, NODE, 0, stream>>>(p, wsB, wsBias);

    conv_wmma<<<dim3(T_LEN / TROWS, B_SZ), 256, 0, stream>>>(
        x, wsB, wsBias, (float*)d_out);
}